// BaeAttn1Decoder_35742717837749
// MI455X (gfx1250) — compile-verified
//
#include <hip/hip_runtime.h>
#include <math.h>

// ---------------------------------------------------------------------------
// BaeAttn1Decoder step for MI455X (gfx1250, wave32, WMMA).
//
// Roofline: after hoisting attn_W out of the S*B projection
// (energies[b,s] = enc[s,b,:] . (attn_W^T h_last[b]) + h_last[b].attn_b),
// the kernel is bound by two streaming passes over encoder_outputs
// (2 x 512 MB) plus ~77 MB of weights => ~1.08 GB => ~46 us @ 23.3 TB/s.
// All dense GEMMs are M=64 tall-skinny f32 and run on V_WMMA_F32_16X16X4_F32.
// ---------------------------------------------------------------------------

typedef float __attribute__((ext_vector_type(2))) v2f;
typedef float __attribute__((ext_vector_type(4))) v4f;
typedef float __attribute__((ext_vector_type(8))) v8f;

#define B_   64
#define H_   1024
#define S_   1024
#define E_   512
#define D2_  2048   // 2*H
#define KX_  2560   // E + 2*H
#define G3_  3072   // 3*H

// ---------------------------------------------------------------------------
// WMMA f32 GEMM: C(64 x N) = A(64 x K) * B (+ bias), f32 throughout.
//   BT = true : Bm is (N x K) row-major (weight matrix W; computes A @ W^T)
//   BT = false: Bm is (K x N) row-major with row stride ldb
// One wave computes a full 64-row x 16-col strip (4 accumulator tiles) so the
// B fragment (the big operand: weights) is loaded from HBM exactly once.
// Fragment layouts per CDNA5 ISA 7.12.2 (f32 A 16x4 / B 4x16 / C 16x16):
//   A: lane<16 -> M=lane, VGPR v = K=k0+v ; lane>=16 -> K=k0+2+v
//   B: lane<16 -> N=lane, VGPR v = K=k0+v ; lane>=16 -> K=k0+2+v
//   C: VGPR j: lanes0-15 M=j, lanes16-31 M=8+j, N=lane%16
// ---------------------------------------------------------------------------
template<bool BT>
__global__ __launch_bounds__(128) void gemm64_wmma(
    const float* __restrict__ A, int K,
    const float* __restrict__ Bm, int ldb,
    const float* __restrict__ bias,
    float* __restrict__ C, int N)
{
  const int lane = threadIdx.x & 31;
  const int wav  = threadIdx.x >> 5;
  const int n0   = (blockIdx.x * 4 + wav) * 16;
  const int half = lane >> 4;       // which K pair this lane holds
  const int lm   = lane & 15;

  v8f c0 = {}, c1 = {}, c2 = {}, c3 = {};

  const float* a0p = A + (size_t)(lm     ) * K;
  const float* a1p = A + (size_t)(lm + 16) * K;
  const float* a2p = A + (size_t)(lm + 32) * K;
  const float* a3p = A + (size_t)(lm + 48) * K;

  for (int k0 = 0; k0 < K; k0 += 4) {
    const int kk = k0 + 2 * half;
    v2f bf;
    if (BT) {
      bf = *(const v2f*)(Bm + (size_t)(n0 + lm) * ldb + kk);
    } else {
      bf.x = Bm[(size_t)(kk    ) * ldb + n0 + lm];
      bf.y = Bm[(size_t)(kk + 1) * ldb + n0 + lm];
    }
    v2f a0 = *(const v2f*)(a0p + kk);
    v2f a1 = *(const v2f*)(a1p + kk);
    v2f a2 = *(const v2f*)(a2p + kk);
    v2f a3 = *(const v2f*)(a3p + kk);
    // 8 args: (neg_a, A, neg_b, B, c_mod, C, reuse_a, reuse_b)
    c0 = __builtin_amdgcn_wmma_f32_16x16x4_f32(false, a0, false, bf, (short)0, c0, false, false);
    c1 = __builtin_amdgcn_wmma_f32_16x16x4_f32(false, a1, false, bf, (short)0, c1, false, false);
    c2 = __builtin_amdgcn_wmma_f32_16x16x4_f32(false, a2, false, bf, (short)0, c2, false, false);
    c3 = __builtin_amdgcn_wmma_f32_16x16x4_f32(false, a3, false, bf, (short)0, c3, false, false);
  }

  const float bv = bias ? bias[n0 + lm] : 0.0f;
  #pragma unroll
  for (int j = 0; j < 8; ++j) {
    const int col = n0 + lm;
    C[(size_t)(     j + 8*half) * N + col] = c0[j] + bv;
    C[(size_t)(16 + j + 8*half) * N + col] = c1[j] + bv;
    C[(size_t)(32 + j + 8*half) * N + col] = c2[j] + bv;
    C[(size_t)(48 + j + 8*half) * N + col] = c3[j] + bv;
  }
}

// bias_dot[b] = h_last[b,:] . attn_b   (one wave per b)
__global__ __launch_bounds__(256) void biasdot_kernel(
    const float* __restrict__ h_last, const float* __restrict__ attn_b,
    float* __restrict__ bias_dot)
{
  const int lane = threadIdx.x & 31;
  const int b = blockIdx.x * 8 + (threadIdx.x >> 5);
  const float* h = h_last + (size_t)b * H_;
  float acc = 0.0f;
  #pragma unroll
  for (int j = 0; j < 8; ++j) {
    v4f hv = *(const v4f*)(h      + j * 128 + lane * 4);
    v4f bb = *(const v4f*)(attn_b + j * 128 + lane * 4);
    v4f p = hv * bb;
    acc += p.x + p.y + p.z + p.w;
  }
  for (int off = 16; off > 0; off >>= 1) acc += __shfl_xor(acc, off, 32);
  if (lane == 0) bias_dot[b] = acc;
}

// energies[b,s] = v[b,:] . enc[s,b,:] + bias_dot[b]  (one wave per (s,b);
// each wave streams an 8 KB contiguous enc row with b128 loads)
__global__ __launch_bounds__(256) void energy_kernel(
    const float* __restrict__ enc, const float* __restrict__ v,
    const float* __restrict__ bias_dot, float* __restrict__ energies)
{
  const int lane = threadIdx.x & 31;
  const int wid  = blockIdx.x * 8 + (threadIdx.x >> 5);
  const int s = wid >> 6;        // / B_
  const int b = wid & (B_ - 1);
  const float* erow = enc + ((size_t)s * B_ + b) * D2_;
  const float* vrow = v + (size_t)b * D2_;
  float acc = 0.0f;
  #pragma unroll
  for (int j = 0; j < 16; ++j) {
    v4f e = *(const v4f*)(erow + j * 128 + lane * 4);
    v4f w = *(const v4f*)(vrow + j * 128 + lane * 4);
    v4f p = e * w;
    acc += p.x + p.y + p.z + p.w;
  }
  for (int off = 16; off > 0; off >>= 1) acc += __shfl_xor(acc, off, 32);
  if (lane == 0) energies[(size_t)b * S_ + s] = acc + bias_dot[b];
}

// attn_w[b,:] = softmax(energies[b,:])   (one block per b)
__global__ __launch_bounds__(256) void softmax_kernel(
    const float* __restrict__ energies, float* __restrict__ attn_out)
{
  __shared__ float red[256];
  const int b = blockIdx.x, t = threadIdx.x;
  const float* row = energies + (size_t)b * S_;
  float vals[4];
  float m = -3.4e38f;
  #pragma unroll
  for (int i = 0; i < 4; ++i) { vals[i] = row[t + i * 256]; m = fmaxf(m, vals[i]); }
  red[t] = m; __syncthreads();
  for (int off = 128; off > 0; off >>= 1) {
    if (t < off) red[t] = fmaxf(red[t], red[t + off]);
    __syncthreads();
  }
  const float mx = red[0]; __syncthreads();
  float s = 0.0f;
  #pragma unroll
  for (int i = 0; i < 4; ++i) { vals[i] = expf(vals[i] - mx); s += vals[i]; }
  red[t] = s; __syncthreads();
  for (int off = 128; off > 0; off >>= 1) {
    if (t < off) red[t] += red[t + off];
    __syncthreads();
  }
  const float inv = 1.0f / red[0];
  #pragma unroll
  for (int i = 0; i < 4; ++i) attn_out[(size_t)b * S_ + t + i * 256] = vals[i] * inv;
}

// x[b, E + d] = sum_s attn_w[b,s] * enc[s,b,d]   (block = (b, 1024-float / 4KB d-chunk);
// attention row cached in LDS; second streaming pass over enc)
__global__ __launch_bounds__(256) void context_kernel(
    const float* __restrict__ enc, const float* __restrict__ attn,
    float* __restrict__ x)
{
  __shared__ float w[S_];
  const int b     = blockIdx.x >> 1;
  const int chunk = blockIdx.x & 1;
  const int t     = threadIdx.x;
  for (int i = t; i < S_; i += 256) w[i] = attn[(size_t)b * S_ + i];
  __syncthreads();
  const int d = chunk * 1024 + t * 4;
  const float* base = enc + (size_t)b * D2_ + d;
  v4f acc = {};
  for (int s = 0; s < S_; s += 4) {
    v4f e0 = *(const v4f*)(base + (size_t)(s    ) * (B_ * D2_));
    v4f e1 = *(const v4f*)(base + (size_t)(s + 1) * (B_ * D2_));
    v4f e2 = *(const v4f*)(base + (size_t)(s + 2) * (B_ * D2_));
    v4f e3 = *(const v4f*)(base + (size_t)(s + 3) * (B_ * D2_));
    acc += w[s] * e0 + w[s + 1] * e1 + w[s + 2] * e2 + w[s + 3] * e3;
  }
  *(v4f*)(x + (size_t)b * KX_ + E_ + d) = acc;
}

// x[b, 0:E] = emb[word[b], :]
__global__ __launch_bounds__(256) void embed_kernel(
    const float* __restrict__ emb, const int* __restrict__ word,
    float* __restrict__ x)
{
  const int idx = blockIdx.x * 256 + threadIdx.x;
  const int b = idx >> 9;          // / E_
  const int e = idx & (E_ - 1);
  x[(size_t)b * KX_ + e] = emb[(size_t)word[b] * E_ + e];
}

__device__ __forceinline__ float sigmoidf_(float v) { return 1.0f / (1.0f + expf(-v)); }

// GRU gate nonlinearity: h = (1-z)*n + z*hprev (gi/gh already include biases)
__global__ __launch_bounds__(256) void gru_act_kernel(
    const float* __restrict__ gi, const float* __restrict__ gh,
    const float* __restrict__ hprev, float* __restrict__ hout,
    float* __restrict__ hout2)
{
  const int idx = blockIdx.x * 256 + threadIdx.x;
  const int b = idx >> 10;
  const int i = idx & (H_ - 1);
  const float* gib = gi + (size_t)b * G3_;
  const float* ghb = gh + (size_t)b * G3_;
  const float r = sigmoidf_(gib[i] + ghb[i]);
  const float z = sigmoidf_(gib[H_ + i] + ghb[H_ + i]);
  const float n = tanhf(gib[2 * H_ + i] + r * ghb[2 * H_ + i]);
  const float h = (1.0f - z) * n + z * hprev[idx];
  hout[idx] = h;
  if (hout2) hout2[idx] = h;
}

extern "C" void kernel_launch(void* const* d_in, const int* in_sizes, int n_in,
                              void* d_out, int out_size, void* d_ws, size_t ws_size,
                              hipStream_t stream)
{
  (void)in_sizes; (void)n_in; (void)out_size; (void)ws_size;

  const int*   word   = (const int*)  d_in[0];
  const float* prev_h = (const float*)d_in[1];   // (2, B, H)
  const float* enc    = (const float*)d_in[2];   // (S, B, 2H)
  const float* emb    = (const float*)d_in[3];   // (VOCAB, E)
  const float* attn_W = (const float*)d_in[4];   // (H, 2H)
  const float* attn_b = (const float*)d_in[5];   // (H,)
  const float* W_ih0  = (const float*)d_in[6];   // (3H, E+2H)
  const float* W_hh0  = (const float*)d_in[7];   // (3H, H)
  const float* b_ih0  = (const float*)d_in[8];
  const float* b_hh0  = (const float*)d_in[9];
  const float* W_ih1  = (const float*)d_in[10];  // (3H, H)
  const float* W_hh1  = (const float*)d_in[11];  // (3H, H)
  const float* b_ih1  = (const float*)d_in[12];
  const float* b_hh1  = (const float*)d_in[13];

  // d_out layout (flat, return order): output (B*H) | hidden (2*B*H) | attn (B*S)
  float* out      = (float*)d_out;
  float* h0_out   = out + B_ * H_;
  float* h1_out   = out + 2 * B_ * H_;
  float* attn_out = out + 3 * B_ * H_;

  // workspace (~4.6 MB of f32)
  float* ws       = (float*)d_ws;
  float* v        = ws;                   // B * 2H
  float* bias_dot = v + B_ * D2_;         // B (rounded to 64)
  float* energies = bias_dot + 64;        // B * S
  float* x        = energies + B_ * S_;   // B * (E + 2H)
  float* gi0      = x + B_ * KX_;         // B * 3H
  float* gh0      = gi0 + B_ * G3_;
  float* gi1      = gh0 + B_ * G3_;
  float* gh1      = gi1 + B_ * G3_;

  const float* h_last = prev_h + (size_t)B_ * H_;   // prev_hidden[-1]

  // 1) v = h_last @ attn_W  (attn_W consumed as (K=H x N=2H), coalesced)
  gemm64_wmma<false><<<D2_ / 64, 128, 0, stream>>>(h_last, H_, attn_W, D2_, nullptr, v, D2_);
  biasdot_kernel<<<B_ / 8, 256, 0, stream>>>(h_last, attn_b, bias_dot);

  // 2) energies = enc . v  (streaming pass 1 over enc, 512 MB)
  energy_kernel<<<(B_ * S_) / 8, 256, 0, stream>>>(enc, v, bias_dot, energies);

  // 3) attn_w = softmax(energies) -> directly into d_out
  softmax_kernel<<<B_, 256, 0, stream>>>(energies, attn_out);

  // 4) x = [emb[word] | context]  (context = streaming pass 2 over enc)
  embed_kernel<<<(B_ * E_) / 256, 256, 0, stream>>>(emb, word, x);
  context_kernel<<<B_ * 2, 256, 0, stream>>>(enc, attn_out, x);

  // 5) GRU layer 0: gates via WMMA GEMMs, then fused nonlinearity
  gemm64_wmma<true><<<G3_ / 64, 128, 0, stream>>>(x,      KX_, W_ih0, KX_, b_ih0, gi0, G3_);
  gemm64_wmma<true><<<G3_ / 64, 128, 0, stream>>>(prev_h, H_,  W_hh0, H_,  b_hh0, gh0, G3_);
  gru_act_kernel<<<(B_ * H_) / 256, 256, 0, stream>>>(gi0, gh0, prev_h, h0_out, nullptr);

  // 6) GRU layer 1 (output h1 duplicated into `output` and `hidden[1]`)
  gemm64_wmma<true><<<G3_ / 64, 128, 0, stream>>>(h0_out, H_, W_ih1, H_, b_ih1, gi1, G3_);
  gemm64_wmma<true><<<G3_ / 64, 128, 0, stream>>>(h_last, H_, W_hh1, H_, b_hh1, gh1, G3_);
  gru_act_kernel<<<(B_ * H_) / 256, 256, 0, stream>>>(gi1, gh1, h_last, h1_out, out);
}